// GraphConvolution_75213467287802
// MI455X (gfx1250) — compile-verified
//
#include <hip/hip_runtime.h>

typedef __attribute__((ext_vector_type(2))) float v2f;
typedef __attribute__((ext_vector_type(8))) float v8f;

#define GC_N_NODES 100000
#define GC_N_EDGES 1600000
#define GC_D 128
#define GC_LDS_STRIDE 132  // 128 + 4 pad -> conflict-free WMMA A-fragment reads

// ---------------------------------------------------------------------------
// Kernel 1: zero the aggregation buffer (d_out doubles as agg, GEMM is in-place)
// ---------------------------------------------------------------------------
__global__ __launch_bounds__(256) void gc_zero_kernel(float4* __restrict__ out, int n4) {
    int i = blockIdx.x * 256 + threadIdx.x;
    if (i < n4) out[i] = make_float4(0.f, 0.f, 0.f, 0.f);
}

// ---------------------------------------------------------------------------
// Kernel 2: edge scatter. One wave32 per edge; lane l handles features 4l..4l+3.
// Gather x[col] (coalesced 512B/wave), scale by edge weight, HW f32 atomic-add
// into agg[row]. agg (51.2MB) is L2-resident (192MB L2) so atomics stay on-chip.
// ---------------------------------------------------------------------------
__global__ __launch_bounds__(256) void gc_scatter_kernel(
    const float* __restrict__ x, const int* __restrict__ erow,
    const int* __restrict__ ecol, const float* __restrict__ evals,
    float* __restrict__ agg) {
    unsigned tid  = blockIdx.x * 256u + threadIdx.x;
    unsigned e    = tid >> 5;   // wave32: one edge per wave
    unsigned lane = tid & 31u;
    if (e >= GC_N_EDGES) return;

    int   r = erow[e];
    int   c = ecol[e];
    float v = evals[e];

    const float4 m = reinterpret_cast<const float4*>(x + (size_t)c * GC_D)[lane];
    float* dst = agg + (size_t)r * GC_D + lane * 4;
    unsafeAtomicAdd(dst + 0, m.x * v);
    unsafeAtomicAdd(dst + 1, m.y * v);
    unsafeAtomicAdd(dst + 2, m.z * v);
    unsafeAtomicAdd(dst + 3, m.w * v);
}

// ---------------------------------------------------------------------------
// Kernel 3: in-place GEMM  out[16 rows] = agg[16 rows] @ W  via
// V_WMMA_F32_16X16X4_F32. Block = 256 threads = 8 waves; each wave owns one
// 16-column band (8 x 16 = 128 columns). A strip staged in LDS before the
// barrier so the in-place overwrite is race-free.
// ---------------------------------------------------------------------------
__global__ __launch_bounds__(256) void gc_gemm_kernel(float* __restrict__ io,
                                                      const float* __restrict__ w) {
    __shared__ float As[16 * GC_LDS_STRIDE];

    const int rowBase = blockIdx.x * 16;
    const int tid     = threadIdx.x;

    // Cooperative load of the 16x128 A strip -> LDS (padded row stride 132).
#pragma unroll
    for (int i = 0; i < 2; ++i) {
        int idx  = (tid + i * 256) * 4;   // element index 0..2044 step 4
        int rrow = idx >> 7;              // /128
        int ccol = idx & 127;
        float4 v = *reinterpret_cast<const float4*>(
            io + (size_t)(rowBase + rrow) * GC_D + ccol);
        float* p = &As[rrow * GC_LDS_STRIDE + ccol];
        p[0] = v.x; p[1] = v.y; p[2] = v.z; p[3] = v.w;
    }
    __syncthreads();  // all global reads of A are done; writes below are safe

    const int wave  = tid >> 5;
    const int lane  = tid & 31;
    const int nbase = wave * 16;
    const int half  = lane >> 4;   // 0: K pair {0,1} lanes, 1: K pair {2,3}
    const int mn    = lane & 15;   // M for A-fragment, N for B-fragment

    // Two accumulators to break the WMMA D->C dependency chain.
    v8f acc0 = {};
    v8f acc1 = {};

#pragma unroll
    for (int k = 0; k < GC_D; k += 8) {
        // ---- step k (A: 16x4, B: 4x16) ----
        v2f a0, b0;
        a0.x = As[mn * GC_LDS_STRIDE + k + 2 * half + 0];
        a0.y = As[mn * GC_LDS_STRIDE + k + 2 * half + 1];
        b0.x = w[(size_t)(k + 2 * half + 0) * GC_D + nbase + mn];
        b0.y = w[(size_t)(k + 2 * half + 1) * GC_D + nbase + mn];
        acc0 = __builtin_amdgcn_wmma_f32_16x16x4_f32(
            false, a0, false, b0, (short)0, acc0, false, false);

        // ---- step k+4 ----
        v2f a1, b1;
        a1.x = As[mn * GC_LDS_STRIDE + k + 4 + 2 * half + 0];
        a1.y = As[mn * GC_LDS_STRIDE + k + 4 + 2 * half + 1];
        b1.x = w[(size_t)(k + 4 + 2 * half + 0) * GC_D + nbase + mn];
        b1.y = w[(size_t)(k + 4 + 2 * half + 1) * GC_D + nbase + mn];
        acc1 = __builtin_amdgcn_wmma_f32_16x16x4_f32(
            false, a1, false, b1, (short)0, acc1, false, false);
    }

    // D layout: VGPR r, lane -> M = r + 8*(lane/16), N = lane%16
#pragma unroll
    for (int r = 0; r < 8; ++r) {
        io[(size_t)(rowBase + r + 8 * half) * GC_D + nbase + mn] = acc0[r] + acc1[r];
    }
}

// ---------------------------------------------------------------------------
extern "C" void kernel_launch(void* const* d_in, const int* in_sizes, int n_in,
                              void* d_out, int out_size, void* d_ws, size_t ws_size,
                              hipStream_t stream) {
    const float* x     = (const float*)d_in[0];
    const int*   erow  = (const int*)d_in[1];
    const int*   ecol  = (const int*)d_in[2];
    const float* evals = (const float*)d_in[3];
    const float* wgt   = (const float*)d_in[4];
    float*       out   = (float*)d_out;

    const int n4 = GC_N_NODES * GC_D / 4;  // 3,200,000 float4
    gc_zero_kernel<<<n4 / 256, 256, 0, stream>>>((float4*)out, n4);

    // 1 wave per edge -> 1.6M * 32 threads / 256 = 200,000 blocks
    gc_scatter_kernel<<<(GC_N_EDGES * 32) / 256, 256, 0, stream>>>(
        x, erow, ecol, evals, out);

    // 16 rows per block -> 6250 blocks, 8 waves each
    gc_gemm_kernel<<<GC_N_NODES / 16, 256, 0, stream>>>(out, wgt);
}